// SparseQK_67723044324148
// MI455X (gfx1250) — compile-verified
//
#include <hip/hip_runtime.h>
#include <math.h>

#define D_MODEL 768
#define D_HID   2048
#define NH      12
#define BATCH   4
#define SEQ     128
#define MROWS   (BATCH * SEQ)          // 512
#define SCORE_N ((size_t)BATCH * NH * SEQ * SEQ)   // 786432

typedef __attribute__((ext_vector_type(2))) float v2f;
typedef __attribute__((ext_vector_type(8))) float v8f;
typedef __attribute__((ext_vector_type(4))) int   v4i;

// ---------------------------------------------------------------------------
// Kernel 1: encoder GEMMs.  Q = relu(x @ Wq + bq), K = relu(x @ Wk + bk)
// M=512, N=2048, K=768.  One wave computes a 16x64 tile via 4 WMMA accums.
// 2048 waves total (1024 per matrix) = 256 blocks of 8 waves.
// ---------------------------------------------------------------------------
__global__ __launch_bounds__(256) void enc_gemm(
    const float* __restrict__ x,
    const float* __restrict__ Wq, const float* __restrict__ Wk,
    const float* __restrict__ bq, const float* __restrict__ bk,
    float* __restrict__ Qbuf, float* __restrict__ Kbuf)
{
    const int tid  = threadIdx.x;
    const int lane = tid & 31;
    const int wave = (blockIdx.x * 256 + tid) >> 5;   // 0..2047
    const int which = wave >> 10;                     // 0 = Q, 1 = K
    const int tile  = wave & 1023;
    const int mt = tile & 31;                         // 32 m-tiles of 16
    const int nt = tile >> 5;                         // 32 n-tiles of 64

    const float* W    = which ? Wk : Wq;
    const float* bias = which ? bk : bq;
    float*       Out  = which ? Kbuf : Qbuf;

    const int lm = lane & 15;
    const int c0 = (lane < 16) ? 0 : 2;   // A: lanes 0-15 hold K=0,1; 16-31 hold K=2,3
    const int m  = mt * 16 + lm;
    const int nb = nt * 64;

    v8f acc[4] = {};
    for (int kk = 0; kk < D_MODEL; kk += 4) {
        // A tile: x[m][kk+c0 .. kk+c0+1]  (8B-aligned contiguous pair)
        v2f a = *(const v2f*)(x + (size_t)m * D_MODEL + kk + c0);
        const float* wrow = W + (size_t)(kk + c0) * D_HID + nb + lm;
#pragma unroll
        for (int j = 0; j < 4; ++j) {
            v2f bb;
            bb.x = wrow[j * 16];            // row kk+c0
            bb.y = wrow[D_HID + j * 16];    // row kk+c0+1
            acc[j] = __builtin_amdgcn_wmma_f32_16x16x4_f32(
                false, a, false, bb, (short)0, acc[j], false, false);
        }
    }

    const int mo = (lane < 16) ? 0 : 8;
#pragma unroll
    for (int j = 0; j < 4; ++j) {
        int n = nb + j * 16 + lm;
        float bv = bias[n];
#pragma unroll
        for (int i = 0; i < 8; ++i) {
            float v = acc[j][i] + bv;
            Out[(size_t)(mt * 16 + i + mo) * D_HID + n] = v > 0.f ? v : 0.f;
        }
    }
}

// ---------------------------------------------------------------------------
// Kernel 2: fused acts -> {feature_fires (NT stores), reg partial, score WMMA}.
// One block per (b,q): 512 blocks x 256 threads (8 waves).
// Wave w owns the 16-key tile k in [16w, 16w+16); h streamed in chunks of 32.
// ---------------------------------------------------------------------------
__global__ __launch_bounds__(256) void qk_score(
    const float* __restrict__ Qbuf, const float* __restrict__ Kbuf,
    const float* __restrict__ Wdec, const float* __restrict__ bdec,
    float* __restrict__ score, int* __restrict__ fires,
    float* __restrict__ partials)
{
    __shared__ float sQ[D_HID];           // 8 KB: full query row
    __shared__ float sK[128][33];         // 16.9 KB: K chunk, pad 33 -> conflict-free
    __shared__ float sW[32 * NH];         // 1.5 KB: W_dec chunk
    __shared__ float sRed[256];

    const int tid  = threadIdx.x;
    const int lane = tid & 31;
    const int wv   = tid >> 5;            // wave -> k-tile id (0..7)
    const int bq   = blockIdx.x;          // b*128 + q
    const int b    = bq >> 7;
    const int q    = bq & 127;

    // Load full Q row for this (b,q)
    for (int i = tid; i < D_HID / 4; i += 256)
        *(float4*)&sQ[i * 4] = *(const float4*)(Qbuf + (size_t)bq * D_HID + i * 4);

    const int kr   = tid >> 1;            // fires pass: key row 0..127
    const int hoff = (tid & 1) * 16;      // half of the 32-h chunk
    const int lm   = lane & 15;
    const int c0   = (lane < 16) ? 0 : 2;

    float reg = 0.f;
    v8f acc = {};

    for (int hc = 0; hc < D_HID; hc += 32) {
        __syncthreads();
        // Stage K chunk: each thread 16 floats (4x float4 from global, scalar to LDS)
        const float* krow = Kbuf + (size_t)(b * SEQ + kr) * D_HID + hc + hoff;
#pragma unroll
        for (int j = 0; j < 4; ++j) {
            float4 kv = *(const float4*)(krow + j * 4);
            sK[kr][hoff + j * 4 + 0] = kv.x;
            sK[kr][hoff + j * 4 + 1] = kv.y;
            sK[kr][hoff + j * 4 + 2] = kv.z;
            sK[kr][hoff + j * 4 + 3] = kv.w;
        }
        // Stage W_dec chunk (rows hc..hc+31, contiguous row-major)
        for (int i = tid; i < 32 * NH; i += 256)
            sW[i] = Wdec[(size_t)hc * NH + i];
        __syncthreads();

        // fires + reg pass: thread covers (k=kr, h in [hc+hoff, hc+hoff+16))
        {
            size_t fbase = ((size_t)bq * SEQ + kr) * (size_t)D_HID + hc + hoff;
            int vals[16];
#pragma unroll
            for (int j = 0; j < 16; ++j) {
                float act = sQ[hc + hoff + j] * sK[kr][hoff + j];
                vals[j] = (act > 0.f) ? 1 : 0;
                reg += sqrtf(act + 1e-6f);     // acts >= 0 post-ReLU
            }
            // fires base float-index = 786433 (== 1 mod 4): 3 scalars, 3x v4i, 1 scalar
            __builtin_nontemporal_store(vals[0], fires + fbase + 0);
            __builtin_nontemporal_store(vals[1], fires + fbase + 1);
            __builtin_nontemporal_store(vals[2], fires + fbase + 2);
            v4i p0 = {vals[3],  vals[4],  vals[5],  vals[6]};
            v4i p1 = {vals[7],  vals[8],  vals[9],  vals[10]};
            v4i p2 = {vals[11], vals[12], vals[13], vals[14]};
            __builtin_nontemporal_store(p0, (v4i*)(fires + fbase + 3));
            __builtin_nontemporal_store(p1, (v4i*)(fires + fbase + 7));
            __builtin_nontemporal_store(p2, (v4i*)(fires + fbase + 11));
            __builtin_nontemporal_store(vals[15], fires + fbase + 15);
        }

        // Score WMMA: A[k_local][h] = Q[h]*K[k][h], B[h][n] = W_dec[h][n]
#pragma unroll
        for (int hh = 0; hh < 32; hh += 4) {
            v2f a, bb;
            a.x = sQ[hc + hh + c0]     * sK[wv * 16 + lm][hh + c0];
            a.y = sQ[hc + hh + c0 + 1] * sK[wv * 16 + lm][hh + c0 + 1];
            bb.x = (lm < NH) ? sW[(hh + c0) * NH + lm]     : 0.f;
            bb.y = (lm < NH) ? sW[(hh + c0 + 1) * NH + lm] : 0.f;
            acc = __builtin_amdgcn_wmma_f32_16x16x4_f32(
                false, a, false, bb, (short)0, acc, false, false);
        }
    }

    // Deterministic block reduction of reg partial
    sRed[tid] = reg;
    __syncthreads();
    for (int st = 128; st > 0; st >>= 1) {
        if (tid < st) sRed[tid] += sRed[tid + st];
        __syncthreads();
    }
    if (tid == 0) partials[blockIdx.x] = sRed[0];

    // Score epilogue: out layout (b, n, q, k); scale 1/sqrt(64) = 0.125
    if (lm < NH) {
        float bd = bdec[lm];
#pragma unroll
        for (int i = 0; i < 8; ++i) {
            int kidx = wv * 16 + i + ((lane < 16) ? 0 : 8);
            size_t idx = (((size_t)b * NH + lm) * SEQ + q) * SEQ + kidx;
            score[idx] = acc[i] * 0.125f + bd;
        }
    }
}

// ---------------------------------------------------------------------------
// Kernel 3: deterministic final reduction of 512 per-block partials.
// ---------------------------------------------------------------------------
__global__ __launch_bounds__(256) void finalize_reg(
    const float* __restrict__ partials, float* __restrict__ out_reg)
{
    __shared__ float s[256];
    s[threadIdx.x] = partials[threadIdx.x] + partials[threadIdx.x + 256];
    __syncthreads();
    for (int st = 128; st > 0; st >>= 1) {
        if (threadIdx.x < st) s[threadIdx.x] += s[threadIdx.x + st];
        __syncthreads();
    }
    if (threadIdx.x == 0) *out_reg = 1e-3f * s[0];
}

// ---------------------------------------------------------------------------
extern "C" void kernel_launch(void* const* d_in, const int* in_sizes, int n_in,
                              void* d_out, int out_size, void* d_ws, size_t ws_size,
                              hipStream_t stream)
{
    (void)in_sizes; (void)n_in; (void)out_size; (void)ws_size;
    const float* x  = (const float*)d_in[0];
    const float* Wq = (const float*)d_in[1];
    const float* Wk = (const float*)d_in[2];
    const float* bq = (const float*)d_in[3];
    const float* bk = (const float*)d_in[4];
    const float* Wd = (const float*)d_in[5];
    const float* bd = (const float*)d_in[6];

    float* Qbuf     = (float*)d_ws;                       // 512*2048 f32 (4 MB)
    float* Kbuf     = Qbuf + (size_t)MROWS * D_HID;       // 4 MB
    float* partials = Kbuf + (size_t)MROWS * D_HID;       // 512 f32

    float* score = (float*)d_out;                         // 786432 f32
    float* regp  = score + SCORE_N;                       // 1 f32
    int*   fires = (int*)(regp + 1);                      // 134217728 i32

    enc_gemm<<<256, 256, 0, stream>>>(x, Wq, Wk, bq, bk, Qbuf, Kbuf);
    qk_score<<<512, 256, 0, stream>>>(Qbuf, Kbuf, Wd, bd, score, fires, partials);
    finalize_reg<<<1, 256, 0, stream>>>(partials, regp);
}